// GCN_44435731644444
// MI455X (gfx1250) — compile-verified
//
#include <hip/hip_runtime.h>
#include <math.h>

typedef float v2f __attribute__((ext_vector_type(2)));
typedef float v8f __attribute__((ext_vector_type(8)));

#define BN_EPS 1e-5f

// ---------------- degree / normalization ----------------

__global__ void k_init_deg(float* deg, int n) {
    int i = blockIdx.x * blockDim.x + threadIdx.x;
    if (i < n) deg[i] = 1.0f;  // self loop contributes 1 to every node's degree
}

__global__ void k_accum_deg(const int* __restrict__ dst, int E, float* deg) {
    int e = blockIdx.x * blockDim.x + threadIdx.x;
    if (e < E) unsafeAtomicAdd(&deg[dst[e]], 1.0f);
}

__global__ void k_deg_to_dinv(float* deg, int n) {
    int i = blockIdx.x * blockDim.x + threadIdx.x;
    if (i < n) {
        float d = deg[i];
        deg[i] = (d > 0.0f) ? rsqrtf(d) : 0.0f;
    }
}

// ---------------- WMMA fp32 GEMM: C[M,N] = A[M,K=128] @ W[K=128,N] ----------------
// One wave32 per 16x16 output tile; 32 x V_WMMA_F32_16X16X4_F32 over K.
// A-operand layout (ISA 7.12.2, 32-bit 16x4): lanes 0-15 = M rows, VGPR0 holds
// K=k0 (lanes 0-15) / K=k0+2 (lanes 16-31), VGPR1 holds K=k0+1 / K=k0+3.
// B-operand mirrored with N across lanes. C/D: VGPR j = row (j + 8*(lane/16)).

template <int N>
__global__ void k_gemm_wmma(const float* __restrict__ A,
                            const float* __restrict__ W,
                            float* __restrict__ C, int M) {
    constexpr int K = 128;
    const int lane = threadIdx.x;      // 0..31
    const int half = lane >> 4;        // 0 or 1
    const int l16  = lane & 15;
    const int m0 = blockIdx.x * 16;
    const int n0 = blockIdx.y * 16;

    int row = m0 + l16;
    if (row >= M) row = M - 1;         // clamp: keeps EXEC all-ones for WMMA
    const float* __restrict__ arow = A + (size_t)row * K;

    v8f acc = {};
#pragma unroll
    for (int k0 = 0; k0 < K; k0 += 4) {
        const int ka = k0 + 2 * half;
        v2f a, b;
        a.x = arow[ka];
        a.y = arow[ka + 1];
        b.x = W[(size_t)ka * N + n0 + l16];
        b.y = W[(size_t)(ka + 1) * N + n0 + l16];
        acc = __builtin_amdgcn_wmma_f32_16x16x4_f32(
            /*neg_a=*/false, a, /*neg_b=*/false, b,
            /*c_mod=*/(short)0, acc, /*reuse_a=*/false, /*reuse_b=*/false);
    }

#pragma unroll
    for (int j = 0; j < 8; ++j) {
        const int m = m0 + j + 8 * half;
        if (m < M) C[(size_t)m * N + n0 + l16] = acc[j];
    }
}

// ---------------- zero fill (float4 stores) ----------------

__global__ void k_zero4(float4* p, size_t n4) {
    size_t i = (size_t)blockIdx.x * blockDim.x + threadIdx.x;
    if (i < n4) p[i] = make_float4(0.f, 0.f, 0.f, 0.f);
}

// ---------------- edge scatter-add: agg[dst] += dinv[src]*dinv[dst]*h[src] ----------------
// One wave32 per edge, NF/32 floats per lane; float4/float2 gather, fp32 HW atomics.

template <int NF>
__global__ void k_edge_agg(const int* __restrict__ src,
                           const int* __restrict__ dst,
                           const float* __restrict__ dinv,
                           const float* __restrict__ h,
                           float* __restrict__ agg, int E) {
    constexpr int VPT = NF / 32;  // 4 for d=128, 2 for d=64
    const int wid  = (int)(((size_t)blockIdx.x * blockDim.x + threadIdx.x) >> 5);
    const int lane = threadIdx.x & 31;
    if (wid >= E) return;

    const int s = src[wid];
    const int d = dst[wid];
    const float nrm = dinv[s] * dinv[d];

    const float* __restrict__ hp = h + (size_t)s * NF + lane * VPT;
    float* __restrict__ ap = agg + (size_t)d * NF + lane * VPT;

#pragma unroll
    for (int t = 0; t < VPT; ++t) {
        unsafeAtomicAdd(&ap[t], nrm * hp[t]);
    }
}

// ---------------- epilogue: self-loop + bias (+ BN + SiLU) ----------------

template <int NF, bool BNACT>
__global__ void k_epilogue(const float* __restrict__ h,
                           const float* __restrict__ dinv,
                           const float* __restrict__ b,
                           const float* __restrict__ g,
                           const float* __restrict__ be,
                           const float* __restrict__ m,
                           const float* __restrict__ var,
                           float* __restrict__ agg_io, int n) {
    size_t idx = (size_t)blockIdx.x * blockDim.x + threadIdx.x;
    const size_t total = (size_t)n * NF;
    if (idx >= total) return;

    const int f = (int)(idx % NF);
    const size_t node = idx / NF;

    const float di = dinv[node];
    float v = agg_io[idx] + di * di * h[idx] + b[f];

    if constexpr (BNACT) {
        v = (v - m[f]) * rsqrtf(var[f] + BN_EPS) * g[f] + be[f];
        v = v / (1.0f + __expf(-v));  // SiLU
    }
    agg_io[idx] = v;
}

// ---------------- host-side orchestration ----------------

extern "C" void kernel_launch(void* const* d_in, const int* in_sizes, int n_in,
                              void* d_out, int out_size, void* d_ws, size_t ws_size,
                              hipStream_t stream) {
    const float* x    = (const float*)d_in[0];
    const int*   ei   = (const int*)d_in[1];   // int32 (JAX x64 disabled)
    const float* W1   = (const float*)d_in[2];
    const float* b1   = (const float*)d_in[3];
    const float* g1   = (const float*)d_in[4];
    const float* be1  = (const float*)d_in[5];
    const float* m1   = (const float*)d_in[6];
    const float* v1   = (const float*)d_in[7];
    const float* W2   = (const float*)d_in[8];
    const float* b2   = (const float*)d_in[9];
    const float* g2   = (const float*)d_in[10];
    const float* be2  = (const float*)d_in[11];
    const float* m2   = (const float*)d_in[12];
    const float* v2   = (const float*)d_in[13];
    const float* W3   = (const float*)d_in[14];
    const float* b3   = (const float*)d_in[15];

    const int n = in_sizes[0] / 128;       // 50000 nodes
    const int E = in_sizes[1] / 2;         // 800000 edges
    const int* srcI = ei;                  // edge_index[0]
    const int* dstI = ei + E;              // edge_index[1]
    float* out = (float*)d_out;

    // workspace layout
    float* ws = (float*)d_ws;
    const size_t nPad = ((size_t)n + 255) & ~(size_t)255;
    float* dinv = ws;                      // n floats
    float* bufA = ws + nPad;               // n*128 floats (dense-transform output)
    float* bufB = bufA + (size_t)n * 128;  // n*128 floats (aggregate / activation)

    const int T = 256;
    const int nb_n = (n + T - 1) / T;
    const int nb_e = (E + T - 1) / T;
    const int mt   = (n + 15) / 16;        // 3125 M-tiles
    const int eb   = (E + 7) / 8;          // 8 waves (edges) per 256-thread block
    const size_t q128 = (size_t)n * 128 / 4;
    const size_t q64  = (size_t)n * 64 / 4;
    const size_t t128 = (size_t)n * 128;
    const size_t t64  = (size_t)n * 64;

    // ---- gcn_norm ----
    k_init_deg<<<nb_n, T, 0, stream>>>(dinv, n);
    k_accum_deg<<<nb_e, T, 0, stream>>>(dstI, E, dinv);
    k_deg_to_dinv<<<nb_n, T, 0, stream>>>(dinv, n);

    // ---- layer 1: h = x@W1 ; agg ; BN+SiLU ----
    k_gemm_wmma<128><<<dim3(mt, 8), 32, 0, stream>>>(x, W1, bufA, n);
    k_zero4<<<(int)((q128 + T - 1) / T), T, 0, stream>>>((float4*)bufB, q128);
    k_edge_agg<128><<<eb, T, 0, stream>>>(srcI, dstI, dinv, bufA, bufB, E);
    k_epilogue<128, true><<<(int)((t128 + T - 1) / T), T, 0, stream>>>(
        bufA, dinv, b1, g1, be1, m1, v1, bufB, n);

    // ---- layer 2 ----
    k_gemm_wmma<128><<<dim3(mt, 8), 32, 0, stream>>>(bufB, W2, bufA, n);
    k_zero4<<<(int)((q128 + T - 1) / T), T, 0, stream>>>((float4*)bufB, q128);
    k_edge_agg<128><<<eb, T, 0, stream>>>(srcI, dstI, dinv, bufA, bufB, E);
    k_epilogue<128, true><<<(int)((t128 + T - 1) / T), T, 0, stream>>>(
        bufA, dinv, b2, g2, be2, m2, v2, bufB, n);

    // ---- layer 3 (d_out = 64, no BN/activation) ----
    k_gemm_wmma<64><<<dim3(mt, 4), 32, 0, stream>>>(bufB, W3, bufA, n);
    k_zero4<<<(int)((q64 + T - 1) / T), T, 0, stream>>>((float4*)out, q64);
    k_edge_agg<64><<<eb, T, 0, stream>>>(srcI, dstI, dinv, bufA, out, E);
    k_epilogue<64, false><<<(int)((t64 + T - 1) / T), T, 0, stream>>>(
        bufA, dinv, b3, nullptr, nullptr, nullptr, nullptr, out, n);
}